// TaskLoss_recon_8899172238075
// MI455X (gfx1250) — compile-verified
//
#include <hip/hip_runtime.h>

// Chamfer distance, B=8, N=4096, D=3, fp32.
// P = aN[i] + bN[j] - 2*A·B^T; the GEMM term uses V_WMMA_F32_16X16X4_F32
// (K padded 3->4 with zeros). Two symmetric passes compute min-over-columns
// for each row of A (A=gt then A=pred). Data is L2-resident (786 KB), so
// plain global loads are the right data path (no TDM staging needed).

typedef __attribute__((ext_vector_type(2))) float v2f;
typedef __attribute__((ext_vector_type(8))) float v8f;

__global__ void zero_out_kernel(float* out) { out[0] = 0.0f; }

// One block = one (batch, 16-row tile of A). 8 waves strip-mine the N/16
// column tiles of B. Each wave computes a 16x16 distance tile per iteration
// with a single f32 WMMA, tracks per-row running mins, then waves combine via
// LDS atomic-min (uint bits; distances clamped >= 0 so ordering is safe).
__global__ __launch_bounds__(256) void chamfer_min_pass(
    const float* __restrict__ Apts,   // [B, N, 3] rows of the distance matrix
    const float* __restrict__ Bpts,   // [B, N, 3] cols of the distance matrix
    float* __restrict__ out,          // scalar accumulator
    int N, float scale)
{
    const int ntiles = N >> 4;
    const int b      = blockIdx.x / ntiles;
    const int itile  = blockIdx.x % ntiles;
    const int tid    = threadIdx.x;
    const int lane   = tid & 31;
    const int hf     = lane >> 4;     // 0: lanes 0-15, 1: lanes 16-31
    const int l15    = lane & 15;
    // Wave index is identical across the wave: pin it to an SGPR so the tile
    // loop gets scalar control flow (s_cbranch, EXEC untouched -> WMMA-legal
    // by construction, no per-iteration exec-mask bookkeeping).
    const int wave   = __builtin_amdgcn_readfirstlane(tid >> 5);

    __shared__ unsigned smin[16];
    if (tid < 16) smin[tid] = 0x7F800000u;   // +inf bits
    __syncthreads();

    const float* __restrict__ Ab = Apts + (size_t)b * N * 3;
    const float* __restrict__ Bb = Bpts + (size_t)b * N * 3;

    // ---- A operand (16x4 f32, MxK): lane holds point M = l15.
    // Lanes 0-15: VGPR0=K0(x), VGPR1=K1(y).  Lanes 16-31: VGPR0=K2(z), VGPR1=K3(0).
    const int am = itile * 16 + l15;
    const float ax = Ab[am * 3 + 0];
    const float ay = Ab[am * 3 + 1];
    const float az = Ab[am * 3 + 2];
    const float aN = ax * ax + ay * ay + az * az;
    v2f avec;
    avec.x = hf ? az : ax;
    avec.y = hf ? 0.0f : ay;

    // Row norms needed by the C/D layout: VGPR r of C holds row (r + 8*hf).
    float rn[8];
#pragma unroll
    for (int r = 0; r < 8; ++r)
        rn[r] = __shfl(aN, hf * 8 + r, 32);

    float vmin[8];
#pragma unroll
    for (int r = 0; r < 8; ++r) vmin[r] = __builtin_huge_valf();

    for (int jt = wave; jt < ntiles; jt += 8) {
        // ---- B operand (4x16 f32, KxN): lane holds column N = l15.
        // VGPR v: lanes 0-15 -> K=v, lanes 16-31 -> K=v+2.
        const int c  = jt * 16 + l15;
        const float bx = Bb[c * 3 + 0];
        const float by = Bb[c * 3 + 1];
        const float bz = Bb[c * 3 + 2];
        const float bN = bx * bx + by * by + bz * bz;
        v2f bvec;
        bvec.x = hf ? bz : bx;
        bvec.y = hf ? 0.0f : by;

        v8f acc = {};
        // D = A x B (+0): dot(A_row, B_col) for a 16x16 tile, one instruction.
        acc = __builtin_amdgcn_wmma_f32_16x16x4_f32(
            /*neg_a=*/false, avec, /*neg_b=*/false, bvec,
            /*c_mod=*/(short)0, acc, /*reuse_a=*/false, /*reuse_b=*/false);

#pragma unroll
        for (int r = 0; r < 8; ++r) {
            float d = rn[r] + bN - 2.0f * acc[r];
            d = fmaxf(d, 0.0f);               // keep >= 0 -> uint-min safe
            vmin[r] = fminf(vmin[r], d);
        }
    }

    // Min across the 16 columns held by this half-wave (xor stays in-half),
    // then combine the 8 waves through LDS.
#pragma unroll
    for (int r = 0; r < 8; ++r) {
        float v = vmin[r];
        v = fminf(v, __shfl_xor(v, 1, 32));
        v = fminf(v, __shfl_xor(v, 2, 32));
        v = fminf(v, __shfl_xor(v, 4, 32));
        v = fminf(v, __shfl_xor(v, 8, 32));
        if (l15 == 0)
            atomicMin(&smin[hf * 8 + r], __float_as_uint(v));
    }
    __syncthreads();

    // Block owns its 16 rows outright: fold into the scalar loss directly.
    if (tid == 0) {
        float s = 0.0f;
#pragma unroll
        for (int r = 0; r < 16; ++r) s += __uint_as_float(smin[r]);
        atomicAdd(out, s * scale);
    }
}

extern "C" void kernel_launch(void* const* d_in, const int* in_sizes, int n_in,
                              void* d_out, int out_size, void* d_ws, size_t ws_size,
                              hipStream_t stream) {
    const float* pred = (const float*)d_in[0];
    const float* gt   = (const float*)d_in[1];
    float* out        = (float*)d_out;

    const int D = 3;
    const int N = 4096;
    const int B = in_sizes[0] / (N * D);   // 8
    const float scale = 1.0f / (float)(B * N);

    zero_out_kernel<<<1, 1, 0, stream>>>(out);

    dim3 grid(B * (N / 16));
    dim3 block(256);
    // loss_2: for each gt row, min over pred columns
    chamfer_min_pass<<<grid, block, 0, stream>>>(gt, pred, out, N, scale);
    // loss_1: for each pred row, min over gt columns
    chamfer_min_pass<<<grid, block, 0, stream>>>(pred, gt, out, N, scale);
}